// chamfer_6DDist_29755533427157
// MI455X (gfx1250) — compile-verified
//
#include <hip/hip_runtime.h>

typedef float v2f __attribute__((ext_vector_type(2)));
typedef float v8f __attribute__((ext_vector_type(8)));

#define DIMS 6
#define TILES 2                                // 16-row tiles per wave
#define ROWS_PER_WAVE (16 * TILES)             // 32
#define WAVES_PER_BLOCK 8
#define ROWS_PER_BLOCK (ROWS_PER_WAVE * WAVES_PER_BLOCK)  // 256

// Fused both chamfer directions (blockIdx.z) in one launch; each wave owns two
// 16-row tiles of P and streams Q in 16-col tiles: one q-row load + one ||q||^2
// feeds 4 x V_WMMA_F32_16X16X4_F32 (K=6 padded to 8 over two K-steps).
__global__ __launch_bounds__(256) void chamfer_nn_fused(
    const float* __restrict__ X1, const float* __restrict__ X2,
    float* __restrict__ dist1, int* __restrict__ idx1,
    float* __restrict__ dist2, int* __restrict__ idx2,
    int N, int M)
{
  const int dir  = blockIdx.z;                 // 0: X1->X2, 1: X2->X1
  const float* P = dir ? X2 : X1;
  const float* Q = dir ? X1 : X2;
  float* dist    = dir ? dist2 : dist1;
  int*   idx     = dir ? idx2  : idx1;
  const int NP   = dir ? M : N;
  const int NQ   = dir ? N : M;

  const int b    = blockIdx.y;
  const int lane = threadIdx.x & 31;
  const int wave = threadIdx.x >> 5;
  const int hi   = lane >> 4;                  // 0: K=0,1 lanes; 1: K=2,3 lanes
  const int lid  = lane & 15;
  const int rowBase = blockIdx.x * ROWS_PER_BLOCK + wave * ROWS_PER_WAVE;

  const float* Pb = P + (size_t)b * NP * DIMS;
  const float* Qb = Q + (size_t)b * NQ * DIMS;

  v2f zz; zz.x = 0.0f; zz.y = 0.0f;

  // ---- loop-invariant A tiles: tile t mirrors row rowBase + 16t + lid ----
  v2f   a0[TILES], a1[TILES];
  float sqrow[TILES];
  float bestD[TILES][8];
  int   bestI[TILES][8];
  #pragma unroll
  for (int t = 0; t < TILES; ++t) {
    const float* prow = Pb + (size_t)(rowBase + 16 * t + lid) * DIMS;
    const v2f p01 = *(const v2f*)(prow + 0);
    const v2f p23 = *(const v2f*)(prow + 2);
    const v2f p45 = *(const v2f*)(prow + 4);
    sqrow[t] = p01.x*p01.x + p01.y*p01.y + p23.x*p23.x + p23.y*p23.y
             + p45.x*p45.x + p45.y*p45.y;
    a0[t] = hi ? p23 : p01;                    // K-step 0: K = 0..3
    a1[t] = hi ? zz  : p45;                    // K-step 1: K = 4..7 (6,7 pad)
    #pragma unroll
    for (int i = 0; i < 8; ++i) { bestD[t][i] = 3.4e38f; bestI[t][i] = 0; }
  }

  int col = lid;                               // this lane's column in C/D
  for (int mBase = 0; mBase < NQ; mBase += 16, col += 16) {
    const float* qrow = Qb + (size_t)(mBase + lid) * DIMS;
    const v2f q01 = *(const v2f*)(qrow + 0);
    const v2f q23 = *(const v2f*)(qrow + 2);
    const v2f q45 = *(const v2f*)(qrow + 4);
    float sq2l = q01.x*q01.x + q01.y*q01.y + q23.x*q23.x + q23.y*q23.y
               + q45.x*q45.x + q45.y*q45.y;
    const v2f b0 = hi ? q23 : q01;
    const v2f b1 = hi ? zz  : q45;

    #pragma unroll
    for (int t = 0; t < TILES; ++t) {
      v8f c = {};
      c = __builtin_amdgcn_wmma_f32_16x16x4_f32(false, a0[t], false, b0,
                                                (short)0, c, false, false);
      c = __builtin_amdgcn_wmma_f32_16x16x4_f32(false, a1[t], false, b1,
                                                (short)0, c, false, false);
      #pragma unroll
      for (int i = 0; i < 8; ++i) {
        float d = __builtin_fmaf(-2.0f, c[i], sq2l);   // ||q||^2 - 2 p.q
        bool better = d < bestD[t][i];
        bestD[t][i] = better ? d   : bestD[t][i];
        bestI[t][i] = better ? col : bestI[t][i];
      }
    }
  }

  // fold in ||p||^2, then (min, argmin) across each 16-lane half; ties -> lower idx
  #pragma unroll
  for (int t = 0; t < TILES; ++t) {
    #pragma unroll
    for (int i = 0; i < 8; ++i) {
      float d  = bestD[t][i] + __shfl(sqrow[t], i + 8 * hi, 16);
      int   ix = bestI[t][i];
      #pragma unroll
      for (int s = 8; s >= 1; s >>= 1) {
        float od = __shfl_xor(d,  s, 16);
        int   oi = __shfl_xor(ix, s, 16);
        if (od < d || (od == d && oi < ix)) { d = od; ix = oi; }
      }
      if (lid == 0) {
        const int r = rowBase + 16 * t + 8 * hi + i;
        dist[(size_t)b * NP + r] = d;
        idx [(size_t)b * NP + r] = ix;
      }
    }
  }
}

extern "C" void kernel_launch(void* const* d_in, const int* in_sizes, int n_in,
                              void* d_out, int out_size, void* d_ws, size_t ws_size,
                              hipStream_t stream) {
  (void)in_sizes; (void)n_in; (void)out_size; (void)d_ws; (void)ws_size;
  const int B = 4, N = 8192, M = 8192;   // N == M required for fused z-grid

  const float* x1 = (const float*)d_in[0];   // [B, N, 6]
  const float* x2 = (const float*)d_in[1];   // [B, M, 6]

  float* o     = (float*)d_out;
  float* dist1 = o;                                   // [B, N]
  float* dist2 = o + (size_t)B * N;                   // [B, M]
  int*   io    = (int*)(o + (size_t)B * N + (size_t)B * M);
  int*   idx1  = io;                                  // [B, N]
  int*   idx2  = io + (size_t)B * N;                  // [B, M]

  dim3 blk(256, 1, 1);
  dim3 grd(N / ROWS_PER_BLOCK, B, 2);     // 32 x 4 x 2 = 256 blocks, 2048 waves

  chamfer_nn_fused<<<grd, blk, 0, stream>>>(x1, x2, dist1, idx1, dist2, idx2, N, M);
}